// GNN_29953101922629
// MI455X (gfx1250) — compile-verified
//
#include <hip/hip_runtime.h>

#define D 128

typedef float v2f __attribute__((ext_vector_type(2)));
typedef float v8f __attribute__((ext_vector_type(8)));
typedef unsigned int u32x4 __attribute__((ext_vector_type(4)));
typedef int i32x4 __attribute__((ext_vector_type(4)));
typedef int i32x8 __attribute__((ext_vector_type(8)));

// ---------------- degree / norm ----------------
__global__ void k_init_deg(float* __restrict__ deg, int n) {
    int i = blockIdx.x * blockDim.x + threadIdx.x;
    if (i < n) deg[i] = 1.0f;  // self-loop contribution
}

__global__ void k_count_deg(const int* __restrict__ dst, float* __restrict__ deg, int e) {
    int i = blockIdx.x * blockDim.x + threadIdx.x;
    if (i < e) atomicAdd(&deg[dst[i]], 1.0f);
}

__global__ void k_dinv(float* __restrict__ deg, int n) {
    int i = blockIdx.x * blockDim.x + threadIdx.x;
    if (i < n) {
        float d = deg[i];
        deg[i] = d > 0.0f ? rsqrtf(d) : 0.0f;  // in-place deg -> dinv
    }
}

// ---------------- dense transform: C[M,128] = A[M,128] * W[128,128] ----------------
// Block = 256 threads = 8 waves. Block stages a 128x128 A tile in LDS (64 KB) via the
// Tensor Data Mover (TDM) when available; wave w computes rows [blockBase+16w, +16)
// x all 128 cols via 8 f32 WMMA tiles (V_WMMA_F32_16X16X4_F32).
__global__ __launch_bounds__(256) void k_gemm_wmma(const float* __restrict__ A,
                                                   const float* __restrict__ W,
                                                   float* __restrict__ C, int M) {
    __shared__ float sA[128 * D];  // 64 KB

    const int tid  = threadIdx.x;
    const int wave = tid >> 5;
    const int lane = tid & 31;
    const int l16  = lane & 15;
    const int half = lane >> 4;  // 0 or 1
    const int blockBase = blockIdx.x * 128;

#if defined(__gfx1250__) && __has_builtin(__builtin_amdgcn_tensor_load_to_lds) && \
    __has_builtin(__builtin_amdgcn_s_wait_tensorcnt)
    // ---- TDM async tile load: one wave issues a 2D tensor DMA into LDS ----
    if (wave == 0) {
        unsigned long long ga = (unsigned long long)(uintptr_t)(A + (size_t)blockBase * D);
        unsigned ldsOff = (unsigned)(uintptr_t)(void*)sA;  // low 32 bits = LDS byte offset
        unsigned dim1 = (unsigned)(M - blockBase);         // rows left: TDM zero-fills OOB

        u32x4 g0;
        g0.x = 1u;                                           // count=1, user descriptor
        g0.y = ldsOff;                                       // lds_addr (bytes)
        g0.z = (unsigned)(ga & 0xFFFFFFFFull);               // global_addr[31:0]
        g0.w = (unsigned)((ga >> 32) & 0x01FFFFFFull) | 0x80000000u;  // addr[56:32] | type=2

        i32x8 g1;
        g1[0] = 0x00020000;                     // data_size=2 (4B); no multicast/pad/iter
        g1[1] = (int)(128u << 16);              // tensor_dim0=128 (low 16 bits)
        g1[2] = (int)((dim1 & 0xFFFFu) << 16);  // tensor_dim0 hi=0 | tensor_dim1 lo
        g1[3] = (int)((dim1 >> 16) | (128u << 16));  // tensor_dim1 hi | tile_dim0=128
        g1[4] = 128;                            // tile_dim1=128, tile_dim2=0
        g1[5] = 128;                            // tensor_dim0_stride=128 (low 32)
        g1[6] = 0;                              // stride hi | tensor_dim1_stride lo
        g1[7] = 0;

        i32x4 gz = (i32x4){0, 0, 0, 0};
#if __clang_major__ >= 23
        i32x8 gz8 = (i32x8){0, 0, 0, 0, 0, 0, 0, 0};
        __builtin_amdgcn_tensor_load_to_lds(g0, g1, gz, gz, gz8, 0);
#else
        __builtin_amdgcn_tensor_load_to_lds(g0, g1, gz, gz, 0);
#endif
        __builtin_amdgcn_s_wait_tensorcnt(0);
    }
    __syncthreads();
#else
    // ---- fallback: coalesced float4 staging ----
    {
        float4* sAv = (float4*)sA;
        for (int i = tid; i < 128 * (D / 4); i += 256) {
            int r   = i >> 5;
            int c4  = i & 31;
            int row = blockBase + r;
            float4 v = make_float4(0.f, 0.f, 0.f, 0.f);
            if (row < M) v = ((const float4*)(A + (size_t)row * D))[c4];
            sAv[i] = v;
        }
    }
    __syncthreads();
#endif

    v8f acc[8];
#pragma unroll
    for (int j = 0; j < 8; ++j) acc[j] = (v8f){0.f, 0.f, 0.f, 0.f, 0.f, 0.f, 0.f, 0.f};

    const float* sArow = sA + (wave * 16 + l16) * D;

    for (int k = 0; k < D; k += 4) {
        const int kk = k + 2 * half;
        v2f afrag;
        afrag.x = sArow[kk + 0];
        afrag.y = sArow[kk + 1];
#pragma unroll
        for (int j = 0; j < 8; ++j) {
            v2f bfrag;
            bfrag.x = W[(size_t)(kk + 0) * D + j * 16 + l16];
            bfrag.y = W[(size_t)(kk + 1) * D + j * 16 + l16];
            acc[j] = __builtin_amdgcn_wmma_f32_16x16x4_f32(
                false, afrag, false, bfrag, (short)0, acc[j], false, false);
        }
    }

    const int rowBase = blockBase + wave * 16;
#pragma unroll
    for (int j = 0; j < 8; ++j) {
#pragma unroll
        for (int v = 0; v < 8; ++v) {
            int row = rowBase + v + 8 * half;
            if (row < M) C[(size_t)row * D + j * 16 + l16] = acc[j][v];
        }
    }
}

// ---------------- self-loop term: agg[i] = dinv[i]^2 * h[i] (initializes agg) -------
__global__ void k_selfloop_init(const float* __restrict__ h, const float* __restrict__ dinv,
                                float* __restrict__ agg, int n) {
    int i = blockIdx.x * blockDim.x + threadIdx.x;  // float4 granularity
    int total = n * (D / 4);
    if (i < total) {
        int node = i >> 5;
        float w = dinv[node];
        w *= w;
        float4 v = ((const float4*)h)[i];
        v.x *= w; v.y *= w; v.z *= w; v.w *= w;
        ((float4*)agg)[i] = v;
    }
}

// ---------------- edge gather-scale-scatter: one wave per edge ----------------
__global__ void k_edge_scatter(const int* __restrict__ src, const int* __restrict__ dst,
                               const float* __restrict__ dinv, const float* __restrict__ H,
                               float* __restrict__ agg, int e) {
    int edge = blockIdx.x * 8 + (threadIdx.x >> 5);
    int lane = threadIdx.x & 31;
    if (edge >= e) return;
    int s = src[edge];
    int d = dst[edge];
    float nrm = dinv[s] * dinv[d];
    float4 m = ((const float4*)(H + (size_t)s * D))[lane];
    float* out = agg + (size_t)d * D + lane * 4;
    atomicAdd(out + 0, m.x * nrm);
    atomicAdd(out + 1, m.y * nrm);
    atomicAdd(out + 2, m.z * nrm);
    atomicAdd(out + 3, m.w * nrm);
}

// ---------------- bias + relu ----------------
__global__ void k_bias_relu(const float* __restrict__ agg, const float* __restrict__ b,
                            float* __restrict__ out, int n) {
    size_t i = (size_t)blockIdx.x * blockDim.x + threadIdx.x;
    if (i < (size_t)n * D) {
        float v = agg[i] + b[i & (D - 1)];
        out[i] = v > 0.f ? v : 0.f;
    }
}

// ---------------- column-mean reduction ----------------
__global__ void k_zero128(float* __restrict__ acc) {
    if (threadIdx.x < D) acc[threadIdx.x] = 0.f;
}

__global__ void k_colsum(const float* __restrict__ agg, float* __restrict__ acc, int n) {
    __shared__ float red[256];
    int col = threadIdx.x & (D - 1);
    int grp = threadIdx.x >> 7;  // 0 or 1
    int row0 = blockIdx.x * 256;
    int rowEnd = row0 + 256 < n ? row0 + 256 : n;
    float s = 0.f;
    for (int r = row0 + grp; r < rowEnd; r += 2) s += agg[(size_t)r * D + col];
    red[threadIdx.x] = s;
    __syncthreads();
    if (threadIdx.x < D) atomicAdd(&acc[col], red[threadIdx.x] + red[threadIdx.x + D]);
}

__global__ void k_finalize(const float* __restrict__ acc, const float* __restrict__ b2,
                           float* __restrict__ out, int n) {
    if (threadIdx.x < D) out[threadIdx.x] = acc[threadIdx.x] / (float)n + b2[threadIdx.x];
}

// ---------------- driver ----------------
extern "C" void kernel_launch(void* const* d_in, const int* in_sizes, int n_in,
                              void* d_out, int out_size, void* d_ws, size_t ws_size,
                              hipStream_t stream) {
    const float* x  = (const float*)d_in[0];
    const int*   ei = (const int*)d_in[1];
    const float* W1 = (const float*)d_in[2];
    const float* b1 = (const float*)d_in[3];
    const float* W2 = (const float*)d_in[4];
    const float* b2 = (const float*)d_in[5];
    float* out = (float*)d_out;

    const int N = in_sizes[0] / D;
    const int E = in_sizes[1] / 2;
    const int* srcIdx = ei;       // edge_index[0]
    const int* dstIdx = ei + E;   // edge_index[1]

    // workspace layout: dinv[N] | acc[128] | B0[N*D] | B1[N*D]
    char* ws = (char*)d_ws;
    float* dinv = (float*)ws;
    size_t off = (((size_t)N * 4) + 511) & ~(size_t)511;
    float* acc = (float*)(ws + off);
    off += 512;
    float* B0 = (float*)(ws + off);
    off += (size_t)N * D * 4;
    float* B1 = (float*)(ws + off);

    dim3 blk(256);
    const int gN    = (N + 255) / 256;
    const int gE    = (E + 255) / 256;
    const int gGemm = (N + 127) / 128;
    const int gSelf = (N * (D / 4) + 255) / 256;
    const int gEdge = (E + 7) / 8;
    const int gND   = (int)(((size_t)N * D + 255) / 256);
    const int gCS   = (N + 255) / 256;

    // normalization
    k_init_deg<<<gN, blk, 0, stream>>>(dinv, N);
    k_count_deg<<<gE, blk, 0, stream>>>(dstIdx, dinv, E);
    k_dinv<<<gN, blk, 0, stream>>>(dinv, N);

    // layer 1: h1 = X*W1 (B0); agg1 = Ahat*h1 (B1); act = relu(agg1+b1) (B0)
    k_gemm_wmma<<<gGemm, blk, 0, stream>>>(x, W1, B0, N);
    k_selfloop_init<<<gSelf, blk, 0, stream>>>(B0, dinv, B1, N);
    k_edge_scatter<<<gEdge, blk, 0, stream>>>(srcIdx, dstIdx, dinv, B0, B1, E);
    k_bias_relu<<<gND, blk, 0, stream>>>(B1, b1, B0, N);

    // layer 2: h2 = act*W2 (B1); agg2 = Ahat*h2 (B0)
    k_gemm_wmma<<<gGemm, blk, 0, stream>>>(B0, W2, B1, N);
    k_selfloop_init<<<gSelf, blk, 0, stream>>>(B1, dinv, B0, N);
    k_edge_scatter<<<gEdge, blk, 0, stream>>>(srcIdx, dstIdx, dinv, B1, B0, E);

    // out = mean_rows(agg2) + b2
    k_zero128<<<1, 128, 0, stream>>>(acc);
    k_colsum<<<gCS, blk, 0, stream>>>(B0, acc, N);
    k_finalize<<<1, 128, 0, stream>>>(acc, b2, out, N);
}